// TCN_18691697672785
// MI455X (gfx1250) — compile-verified
//
#include <hip/hip_runtime.h>

typedef float v2f __attribute__((ext_vector_type(2)));
typedef float v8f __attribute__((ext_vector_type(8)));

#define C_CH  512
#define S_LEN 4096
#define BATCH 16
#define KW    3
#define MT    64       // workgroup M tile (Cout)
#define NT    128      // workgroup N tile (S)
#define NW    64       // per-wave N tile (4 x 16 subtiles)
#define KC    16       // Cin chunk staged in LDS
#define XS_COLS   (NT + KW - 1)   // 130 valid columns
#define XS_STRIDE 132             // padded row stride
#define W_STRIDE  18              // KC + pad (even -> 8B-aligned v2f loads)

__launch_bounds__(256, 2)
__global__ void tcn_layer_kernel(const float* __restrict__ X,
                                 float* __restrict__ Y,
                                 const float* __restrict__ Wr,
                                 const float* __restrict__ Wf,
                                 const float* __restrict__ Wg)
{
    __shared__ float lds[KC*XS_STRIDE + 7*MT*W_STRIDE];
    float* Xs  = lds;                        // [KC][XS_STRIDE]
    float* WrS = Xs  + KC*XS_STRIDE;         // [MT][W_STRIDE]
    float* WfS = WrS + MT*W_STRIDE;          // [KW][MT][W_STRIDE]
    float* WgS = WfS + KW*MT*W_STRIDE;       // [KW][MT][W_STRIDE]

    const int tid  = threadIdx.x;
    const int lane = tid & 31;
    const int w    = tid >> 5;
    const int hl   = lane >> 4;      // lane half: selects K pair / M+8
    const int l15  = lane & 15;
    const int m_woff = (w >> 1) * 16;    // 4 waves across M
    const int n_woff = (w & 1) * NW;     // 2 waves across N, 64 cols each

    const int n_base = blockIdx.x * NT;
    const int m_base = blockIdx.y * MT;
    const long bofs  = (long)blockIdx.z * C_CH * S_LEN;

    const float* Xb = X + bofs;
    float*       Yb = Y + bofs;

    v8f accR[4] = {}; v8f accF[4] = {}; v8f accG[4] = {};

    for (int ci0 = 0; ci0 < C_CH; ci0 += KC) {
        // ---- stage X tile with causal left halo (zeros for s < 0) ----
        for (int idx = tid; idx < KC*XS_COLS; idx += 256) {
            int r = idx / XS_COLS, c = idx % XS_COLS;
            int s = n_base - (KW-1) + c;
            Xs[r*XS_STRIDE + c] = (s >= 0) ? Xb[(long)(ci0 + r)*S_LEN + s] : 0.0f;
        }
        // ---- residual 1x1 weights: [co][ci] ----
        for (int idx = tid; idx < MT*KC; idx += 256) {
            int co = idx / KC, ci = idx % KC;
            WrS[co*W_STRIDE + ci] = Wr[(long)(m_base+co)*C_CH + ci0 + ci];
        }
        // ---- gated conv weights: global [co][ci][k] -> LDS [k][co][ci] ----
        for (int idx = tid; idx < MT*KC*KW; idx += 256) {
            int k  = idx % KW;
            int ci = (idx / KW) % KC;
            int co = idx / (KW*KC);
            long g = ((long)(m_base+co)*C_CH + (ci0 + ci))*KW + k;
            WfS[(k*MT + co)*W_STRIDE + ci] = Wf[g];
            WgS[(k*MT + co)*W_STRIDE + ci] = Wg[g];
        }
        __syncthreads();

        #pragma unroll
        for (int kk = 0; kk < KC/4; ++kk) {
            const int cil = kk*4 + 2*hl;   // this lane's K-pair within chunk

            // A fragments: row = co (lane 0..15), cols = K pair; reused 4x
            const int arow = (m_woff + l15)*W_STRIDE + cil;
            v2f ar  = *(const v2f*)&WrS[arow];
            v2f af0 = *(const v2f*)&WfS[(0*MT)*W_STRIDE + arow];
            v2f af1 = *(const v2f*)&WfS[(1*MT)*W_STRIDE + arow];
            v2f af2 = *(const v2f*)&WfS[(2*MT)*W_STRIDE + arow];
            v2f ag0 = *(const v2f*)&WgS[(0*MT)*W_STRIDE + arow];
            v2f ag1 = *(const v2f*)&WgS[(1*MT)*W_STRIDE + arow];
            v2f ag2 = *(const v2f*)&WgS[(2*MT)*W_STRIDE + arow];

            const float* p0 = &Xs[(cil  )*XS_STRIDE];
            const float* p1 = &Xs[(cil+1)*XS_STRIDE];

            #pragma unroll
            for (int j = 0; j < 4; ++j) {
                // B fragments: rows cil, cil+1 of Xs at the three causal
                // shifts. Output col nn needs X[s - 2 + k] -> Xs col (nn + k).
                const int c = n_woff + 16*j + l15;
                v2f b0, b1, b2;
                b0.x = p0[c  ]; b0.y = p1[c  ];
                b1.x = p0[c+1]; b1.y = p1[c+1];
                b2.x = p0[c+2]; b2.y = p1[c+2];

                // residual 1x1 conv sits at shift k=2 (no look-back)
                accR[j] = __builtin_amdgcn_wmma_f32_16x16x4_f32(false, ar,  false, b2, (short)0, accR[j], false, false);
                accF[j] = __builtin_amdgcn_wmma_f32_16x16x4_f32(false, af0, false, b0, (short)0, accF[j], false, false);
                accF[j] = __builtin_amdgcn_wmma_f32_16x16x4_f32(false, af1, false, b1, (short)0, accF[j], false, false);
                accF[j] = __builtin_amdgcn_wmma_f32_16x16x4_f32(false, af2, false, b2, (short)0, accF[j], false, false);
                accG[j] = __builtin_amdgcn_wmma_f32_16x16x4_f32(false, ag0, false, b0, (short)0, accG[j], false, false);
                accG[j] = __builtin_amdgcn_wmma_f32_16x16x4_f32(false, ag1, false, b1, (short)0, accG[j], false, false);
                accG[j] = __builtin_amdgcn_wmma_f32_16x16x4_f32(false, ag2, false, b2, (short)0, accG[j], false, false);
            }
        }
        __syncthreads();
    }

    // ---- epilogue: tanh(f)*sigmoid(g) + residual; C/D layout VGPR e:
    //      lanes 0-15 -> M=e, lanes 16-31 -> M=8+e; N = lane&15 ----
    const int rbase = m_base + m_woff + 8*hl;
    #pragma unroll
    for (int j = 0; j < 4; ++j) {
        const int col = n_base + n_woff + 16*j + l15;
        #pragma unroll
        for (int e = 0; e < 8; ++e) {
            float f = tanhf(accF[j][e]);
            float g = 1.0f / (1.0f + __expf(-accG[j][e]));
            Yb[(long)(rbase + e)*S_LEN + col] = f*g + accR[j][e];
        }
    }
}

extern "C" void kernel_launch(void* const* d_in, const int* in_sizes, int n_in,
                              void* d_out, int out_size, void* d_ws, size_t ws_size,
                              hipStream_t stream) {
    const float* x  = (const float*)d_in[0];
    const float* Wr = (const float*)d_in[1];   // [L, C, C, 1]
    const float* Wf = (const float*)d_in[2];   // [L, C, C, K]
    const float* Wg = (const float*)d_in[3];
    float* out = (float*)d_out;
    float* tmp = (float*)d_ws;                 // layer-0 activations (134 MB)

    dim3 grid(S_LEN / NT, C_CH / MT, BATCH);
    dim3 block(256);

    const long wr_l = (long)C_CH * C_CH;       // per-layer Wr stride
    const long wf_l = (long)C_CH * C_CH * KW;  // per-layer Wf/Wg stride

    // layer 0: x -> tmp
    tcn_layer_kernel<<<grid, block, 0, stream>>>(x, tmp, Wr, Wf, Wg);
    // layer 1: tmp -> out
    tcn_layer_kernel<<<grid, block, 0, stream>>>(tmp, out, Wr + wr_l, Wf + wf_l, Wg + wf_l);
}